// WeightedAverage_color_temporal_22909355556927
// MI455X (gfx1250) — compile-verified
//
#include <hip/hip_runtime.h>
#include <stdint.h>

// Problem constants from the reference: b=4, n=8, 3 Lab ch, 2 ab ch, 384x384.
#define NFR 8
#define HW  (384 * 384)
#define TPB 256

typedef float v2f __attribute__((ext_vector_type(2)));

__global__ __launch_bounds__(TPB) void wavg_color_temporal_kernel(
    const float* __restrict__ xlab,   // (B, 8, 3, H, W)
    const float* __restrict__ xab,    // (B, 8, 2, H, W)
    const int*   __restrict__ alphaPtr,
    float*       __restrict__ out)    // (B*8, 2, H, W)
{
    const int tid = threadIdx.x;
    const int p   = blockIdx.x * TPB + tid;   // pixel index within (H*W)
    const int b   = blockIdx.y;               // batch index

    // LDS staging for the 16 x_ab streams: [stream][tid] is conflict-free for
    // both the async writes (consecutive dwords per instruction) and readback.
    __shared__ float abStage[2 * NFR][TPB];

    // ---------------- Phase 0: async global -> LDS copy of x_ab -------------
    // CDNA5 async path (ASYNCcnt): overlaps the 16 ab loads with the gram /
    // softmax compute below and keeps their destinations out of VGPRs.
    {
        unsigned gbyte = (unsigned)((b * (NFR * 2 * HW) + p) * 4);
        unsigned lbyte = (unsigned)(uintptr_t)(&abStage[0][tid]); // low 32b = LDS addr
        #pragma unroll
        for (int k = 0; k < 2 * NFR; ++k) {
            unsigned g = gbyte + (unsigned)(k * HW * 4);
            unsigned l = lbyte + (unsigned)(k * TPB * 4);
            asm volatile("global_load_async_to_lds_b32 %0, %1, %2"
                         :: "v"(l), "v"(g), "s"(xab)
                         : "memory");
        }
    }

    // ---------------- Phase 1: load Lab, pairwise weights -------------------
    // L channel kept scalar; a,b channels packed (v_pk_* f32 ops).
    const float* labBase = xlab + ((size_t)b * (NFR * 3 * HW) + p);
    float lL[NFR];
    v2f   lab[NFR];
    #pragma unroll
    for (int n = 0; n < NFR; ++n) {
        lL[n]    = __builtin_nontemporal_load(labBase + (n * 3 + 0) * HW);
        lab[n].x = __builtin_nontemporal_load(labBase + (n * 3 + 1) * HW);
        lab[n].y = __builtin_nontemporal_load(labBase + (n * 3 + 2) * HW);
    }

    // w_ij = exp(-d2/alpha) = exp2(d2 * (-log2e/alpha)); symmetric, diag = 1.
    // -d2/alpha <= 0 so no max-subtraction is needed (denominator >= 1).
    const float negScale =
        -1.4426950408889634f * __builtin_amdgcn_rcpf((float)alphaPtr[0]);

    float wgt[NFR][NFR];   // only upper triangle materializes (full unroll)
    float rsum[NFR];
    #pragma unroll
    for (int i = 0; i < NFR; ++i) rsum[i] = 1.0f;   // diagonal exp(0)

    #pragma unroll
    for (int i = 0; i < NFR; ++i) {
        #pragma unroll
        for (int j = i + 1; j < NFR; ++j) {
            float dL  = lL[i] - lL[j];          // v_sub_f32
            v2f   dab = lab[i] - lab[j];        // v_pk_add_f32 (neg)
            v2f   sq  = dab * dab;              // v_pk_mul_f32
            float dd  = __builtin_fmaf(dL, dL, sq.x + sq.y);
            float e   = __builtin_amdgcn_exp2f(dd * negScale);  // v_exp_f32
            wgt[i][j] = e;
            rsum[i] += e;
            rsum[j] += e;
        }
    }

    // ---------------- Phase 2: wait for async copies, weighted average ------
    asm volatile("s_wait_asynccnt 0" ::: "memory");

    v2f ab[NFR];
    #pragma unroll
    for (int n = 0; n < NFR; ++n) {
        ab[n].x = abStage[2 * n + 0][tid];
        ab[n].y = abStage[2 * n + 1][tid];
    }

    float* outBase = out + ((size_t)b * (NFR * 2 * HW) + p);
    #pragma unroll
    for (int i = 0; i < NFR; ++i) {
        v2f acc = ab[i];                       // j == i term, weight 1
        #pragma unroll
        for (int j = 0; j < NFR; ++j) {
            if (j == i) continue;
            float e = (j > i) ? wgt[i][j] : wgt[j][i];
            v2f ev; ev.x = e; ev.y = e;
            acc = __builtin_elementwise_fma(ev, ab[j], acc);  // v_pk_fma_f32
        }
        float inv = __builtin_amdgcn_rcpf(rsum[i]);           // rsum in [1,8]
        v2f iv; iv.x = inv; iv.y = inv;
        v2f res = acc * iv;                                   // v_pk_mul_f32
        __builtin_nontemporal_store(res.x, outBase + (2 * i + 0) * HW);
        __builtin_nontemporal_store(res.y, outBase + (2 * i + 1) * HW);
    }
}

extern "C" void kernel_launch(void* const* d_in, const int* in_sizes, int n_in,
                              void* d_out, int out_size, void* d_ws, size_t ws_size,
                              hipStream_t stream) {
    const float* xlab  = (const float*)d_in[0];
    const float* xab   = (const float*)d_in[1];
    const int*   alpha = (const int*)d_in[2];
    float*       outp  = (float*)d_out;

    const int B = in_sizes[0] / (NFR * 3 * HW);   // = 4
    dim3 grid(HW / TPB, B);                        // 576 x 4 blocks, exact fit
    hipLaunchKernelGGL(wavg_color_temporal_kernel, grid, dim3(TPB), 0, stream,
                       xlab, xab, alpha, outp);
}